// InvariantCrossAttention_39582418600293
// MI455X (gfx1250) — compile-verified
//
#include <hip/hip_runtime.h>

// ---------------------------------------------------------------------------
// InvariantCrossAttention collapses analytically:
//   softmax_j(-(Q2_i + K2_j)) == softmax_j(-K2_j)   (shift invariance in j)
// so context[b,i] is constant over i and the output is
//   out[b] = ( Σ_j w_j * K2_j ) / ( Σ_j w_j ),  w_j = exp(-(K2_j - min K2))
// with K2_j = (x_j - mean_j x)^2 over all_atom_features.  cdr3_features is
// mathematically irrelevant.  Total traffic: 128 KB -> launch-latency bound.
// Wave-level sums are done on the matrix pipe (v_wmma_f32_16x16x4_f32).
// ---------------------------------------------------------------------------

typedef __attribute__((ext_vector_type(2))) float v2f;
typedef __attribute__((ext_vector_type(8))) float v8f;

// Exact full-wave32 f32 sum using V_WMMA_F32_16X16X4_F32 with an all-ones B.
// A layout (ISA 7.12.2, 32-bit A 16x4): lane L<16 holds A[L,0]=a.x, A[L,1]=a.y;
// lane L>=16 holds A[L-16,2]=a.x, A[L-16,3]=a.y.  With a=(p,0) and B==1:
//   D[m,n] = p[m] + p[m+16]  (for all n)
// D layout: VGPR r, lanes 0-15 -> D[r,lane]; lanes 16-31 -> D[r+8,lane-16].
// Sum of the 8 D VGPRs gives the half-wave partial; shfl_xor(16) finishes.
__device__ __forceinline__ float wave32_sum_wmma(float p) {
    v2f a; a.x = p;   a.y = 0.0f;
    v2f b; b.x = 1.0f; b.y = 1.0f;
    v8f c = {};
    v8f d = __builtin_amdgcn_wmma_f32_16x16x4_f32(
        /*neg_a=*/false, a, /*neg_b=*/false, b,
        /*c_mod=*/(short)0, c, /*reuse_a=*/false, /*reuse_b=*/false);
    float s01 = d[0] + d[1];
    float s23 = d[2] + d[3];
    float s45 = d[4] + d[5];
    float s67 = d[6] + d[7];
    float s = (s01 + s23) + (s45 + s67);
    s += __shfl_xor(s, 16, 32);
    return s;   // full wave sum, broadcast to all 32 lanes
}

__device__ __forceinline__ float wave32_min(float p) {
    #pragma unroll
    for (int off = 16; off >= 1; off >>= 1)
        p = fminf(p, __shfl_xor(p, off, 32));
    return p;
}

#define TPB 256
#define VPT 8           // float4 vectors per thread: 8192 / (256*4)
#define NWAVES (TPB / 32)

__global__ __launch_bounds__(TPB)
void InvariantCrossAttention_39582418600293_kernel(const float* __restrict__ allatom,
                                                   float* __restrict__ out,
                                                   int M) {
    const int b    = blockIdx.x;
    const int t    = threadIdx.x;
    const int wave = t >> 5;
    const int lane = t & 31;
    const int M4   = M >> 2;                          // float4 count (M % 4 == 0)
    const float4* __restrict__ x4 = (const float4*)(allatom + (size_t)b * (size_t)M);

    __shared__ float red_sum[NWAVES];
    __shared__ float red_min[NWAVES];
    __shared__ float red_S[NWAVES];
    __shared__ float red_T[NWAVES];

    // ---- Pass A: load once (global_load_b128), compute mean ----------------
    float4 v[VPT];
    float acc0 = 0.0f, acc1 = 0.0f, acc2 = 0.0f, acc3 = 0.0f;
    #pragma unroll
    for (int i = 0; i < VPT; ++i) {
        int idx = t + i * TPB;
        v[i] = (idx < M4) ? x4[idx] : make_float4(0.f, 0.f, 0.f, 0.f);
        acc0 += v[i].x; acc1 += v[i].y; acc2 += v[i].z; acc3 += v[i].w;
    }
    float psum = (acc0 + acc1) + (acc2 + acc3);
    float wsum = wave32_sum_wmma(psum);               // v_wmma_f32_16x16x4_f32
    if (lane == 0) red_sum[wave] = wsum;
    __syncthreads();
    float total = 0.0f;
    #pragma unroll
    for (int w = 0; w < NWAVES; ++w) total += red_sum[w];
    const float mean = total / (float)M;

    // ---- Pass B: K2 = (x-mean)^2 and its minimum (softmax shift) -----------
    float k2[VPT][4];
    float pmin = 3.402823466e38f;
    #pragma unroll
    for (int i = 0; i < VPT; ++i) {
        int idx = t + i * TPB;
        float d0 = v[i].x - mean, d1 = v[i].y - mean;
        float d2 = v[i].z - mean, d3 = v[i].w - mean;
        k2[i][0] = d0 * d0; k2[i][1] = d1 * d1;
        k2[i][2] = d2 * d2; k2[i][3] = d3 * d3;
        if (idx < M4) {
            float m01 = fminf(k2[i][0], k2[i][1]);
            float m23 = fminf(k2[i][2], k2[i][3]);
            pmin = fminf(pmin, fminf(m01, m23));
        }
    }
    pmin = wave32_min(pmin);
    if (lane == 0) red_min[wave] = pmin;
    __syncthreads();
    float gmin = red_min[0];
    #pragma unroll
    for (int w = 1; w < NWAVES; ++w) gmin = fminf(gmin, red_min[w]);

    // ---- Pass C: S = Σ exp(-(K2-gmin)),  T = Σ exp(-(K2-gmin))*K2 ----------
    float s0 = 0.f, s1 = 0.f, s2 = 0.f, s3 = 0.f;
    float t0 = 0.f, t1 = 0.f, t2 = 0.f, t3 = 0.f;
    #pragma unroll
    for (int i = 0; i < VPT; ++i) {
        int idx = t + i * TPB;
        if (idx < M4) {
            float w0 = __expf(gmin - k2[i][0]);       // v_exp_f32 (TRANS pipe)
            float w1 = __expf(gmin - k2[i][1]);
            float w2 = __expf(gmin - k2[i][2]);
            float w3 = __expf(gmin - k2[i][3]);
            s0 += w0; s1 += w1; s2 += w2; s3 += w3;
            t0 = fmaf(w0, k2[i][0], t0);
            t1 = fmaf(w1, k2[i][1], t1);
            t2 = fmaf(w2, k2[i][2], t2);
            t3 = fmaf(w3, k2[i][3], t3);
        }
    }
    float pS = (s0 + s1) + (s2 + s3);
    float pT = (t0 + t1) + (t2 + t3);
    float Sw = wave32_sum_wmma(pS);                   // v_wmma_f32_16x16x4_f32
    float Tw = wave32_sum_wmma(pT);                   // v_wmma_f32_16x16x4_f32
    if (lane == 0) { red_S[wave] = Sw; red_T[wave] = Tw; }
    __syncthreads();

    if (t == 0) {
        float S = 0.0f, T = 0.0f;
        #pragma unroll
        for (int w = 0; w < NWAVES; ++w) { S += red_S[w]; T += red_T[w]; }
        out[b] = T / S;
    }
}

extern "C" void kernel_launch(void* const* d_in, const int* in_sizes, int n_in,
                              void* d_out, int out_size, void* d_ws, size_t ws_size,
                              hipStream_t stream) {
    // d_in[0] = cdr3_features     [B,N,1]  (analytically unused)
    // d_in[1] = all_atom_features [B,M,1]  float32
    (void)d_ws; (void)ws_size; (void)n_in;
    const float* allatom = (const float*)d_in[1];
    float* out = (float*)d_out;
    const int B = out_size;                 // output shape [B,1]
    const int M = in_sizes[1] / B;          // 8192
    InvariantCrossAttention_39582418600293_kernel<<<dim3(B), dim3(TPB), 0, stream>>>(
        allatom, out, M);
}